// SSIMLoss_26130581029243
// MI455X (gfx1250) — compile-verified
//
#include <hip/hip_runtime.h>

// SSIM loss via separable Gaussian conv expressed as banded matmuls on the
// CDNA5 matrix pipe (V_WMMA_F32_16X16X4_F32), fp32 end-to-end.
// Round 2: branchless zero-padding with clamped, 8B-aligned b64 loads
// (band window shifted by +1 so per-lane fragment columns are even).

typedef float v2f __attribute__((ext_vector_type(2)));
typedef float v8f __attribute__((ext_vector_type(8)));

#define IMG_N 32
#define IMG_H 512
#define IMG_W 512
#define TILES_PER_IMG ((IMG_H / 16) * (IMG_W / 16))   // 1024
#define NTILES (IMG_N * TILES_PER_IMG)                // 32768

// Normalized 11-tap Gaussian, sigma = 1.5
__constant__ float c_gauss[11] = {
    0.0010284f, 0.0075988f, 0.0360008f, 0.1093607f, 0.2130057f,
    0.2660117f, 0.2130057f, 0.1093607f, 0.0360008f, 0.0075988f, 0.0010284f};

// Shifted band matrix: G[m,k] = g[k-1-m]; K index k corresponds to source
// offset (k-6) relative to the tile origin, so k=0 and k=27 rows are zero.
__device__ __forceinline__ float band(int m, int k) {
  int t = k - 1 - m;
  return ((unsigned)t <= 10u) ? c_gauss[t] : 0.0f;
}

__device__ __forceinline__ v8f wmma_f32(v2f a, v2f b, v8f c) {
  // D = A(16x4) * B(4x16) + C, fp32 everywhere.
  return __builtin_amdgcn_wmma_f32_16x16x4_f32(
      /*neg_a=*/false, a, /*neg_b=*/false, b,
      /*c_mod=*/(short)0, c, /*reuse_a=*/false, /*reuse_b=*/false);
}

__global__ __launch_bounds__(32) void ssim_tile_kernel(
    const float* __restrict__ img1, const float* __restrict__ img2,
    float* __restrict__ partials) {
  // Horizontal-pass intermediate: 5 signals x 32 rows x 16 cols (pad to 17).
  __shared__ float Hs[5][32][17];

  const int b = blockIdx.x;
  const int img = b >> 10;          // / TILES_PER_IMG
  const int tid = b & 1023;
  const int ty = tid >> 5;
  const int tx = tid & 31;
  const int r0 = ty * 16;
  const int c0 = tx * 16;

  const float* I1 = img1 + (size_t)img * (IMG_H * IMG_W);
  const float* I2 = img2 + (size_t)img * (IMG_H * IMG_W);

  const int lane = (int)threadIdx.x;
  const int hi = lane >> 4;    // half-wave select (K +2/+3 slots)
  const int mr = lane & 15;    // M (A) / N (B,C,D) index

  // Constant band fragments. The per-lane expression is identical for the
  // A-layout (vertical pass, M=mr) and the B-layout (horizontal pass, N=mr).
  v2f W[7];
#pragma unroll
  for (int c = 0; c < 7; ++c) {
    int k0 = 4 * c + 2 * hi;
    W[c].x = band(mr, k0);
    W[c].y = band(mr, k0 + 1);
  }

  // ---- Horizontal pass: H = In[16x28] x G^T[28x16], two 16-row blocks ----
#pragma unroll
  for (int blk = 0; blk < 2; ++blk) {
    v8f a1 = {}, a2 = {}, a11 = {}, a22 = {}, a12 = {};
    const int row = r0 - 6 + blk * 16 + mr;          // lane's A-fragment row
    const int rowc = min(max(row, 0), IMG_H - 1);    // clamped (never faults)
    const bool rok = (row == rowc);
    const float* p1 = I1 + (size_t)rowc * IMG_W;
    const float* p2 = I2 + (size_t)rowc * IMG_W;
#pragma unroll
    for (int c = 0; c < 7; ++c) {
      const int col = c0 - 6 + 4 * c + 2 * hi;       // even by construction
      const int colc = min(max(col, 0), IMG_W - 2);  // even -> 8B aligned
      const bool ok = rok && ((unsigned)col < (unsigned)IMG_W);
      const v2f d1 = *(const v2f*)(p1 + colc);       // one global_load_b64
      const v2f d2 = *(const v2f*)(p2 + colc);
      const float x0 = ok ? d1.x : 0.0f;             // branchless zero-pad
      const float x1 = ok ? d1.y : 0.0f;
      const float y0 = ok ? d2.x : 0.0f;
      const float y1 = ok ? d2.y : 0.0f;
      v2f A1 = {x0, x1};
      v2f A2 = {y0, y1};
      a1  = wmma_f32(A1,      W[c], a1);
      a2  = wmma_f32(A2,      W[c], a2);
      a11 = wmma_f32(A1 * A1, W[c], a11);
      a22 = wmma_f32(A2 * A2, W[c], a22);
      a12 = wmma_f32(A1 * A2, W[c], a12);
    }
    // C/D layout: VGPR i -> M = i (+8 for upper half-wave), N = mr.
#pragma unroll
    for (int i = 0; i < 8; ++i) {
      const int rr = blk * 16 + i + hi * 8;
      Hs[0][rr][mr] = a1[i];
      Hs[1][rr][mr] = a2[i];
      Hs[2][rr][mr] = a11[i];
      Hs[3][rr][mr] = a22[i];
      Hs[4][rr][mr] = a12[i];
    }
  }
  __syncthreads();

  // ---- Vertical pass: Out = G[16x28] x H[28x16] ----
  v8f m1 = {}, m2 = {}, s11 = {}, s22 = {}, s12 = {};
#pragma unroll
  for (int c = 0; c < 7; ++c) {
    const int kr = 4 * c + 2 * hi;   // B rows for this lane: kr, kr+1 (<= 27)
    v2f B1  = {Hs[0][kr][mr], Hs[0][kr + 1][mr]};
    v2f B2  = {Hs[1][kr][mr], Hs[1][kr + 1][mr]};
    v2f B11 = {Hs[2][kr][mr], Hs[2][kr + 1][mr]};
    v2f B22 = {Hs[3][kr][mr], Hs[3][kr + 1][mr]};
    v2f B12 = {Hs[4][kr][mr], Hs[4][kr + 1][mr]};
    m1  = wmma_f32(W[c], B1,  m1);
    m2  = wmma_f32(W[c], B2,  m2);
    s11 = wmma_f32(W[c], B11, s11);
    s22 = wmma_f32(W[c], B22, s22);
    s12 = wmma_f32(W[c], B12, s12);
  }

  // ---- Elementwise SSIM map + partial sum (8 output pixels per lane) ----
  const float K1 = 0.0001f;   // C1 = 0.01^2
  const float K2 = 0.0009f;   // C2 = 0.03^2
  float part = 0.0f;
#pragma unroll
  for (int i = 0; i < 8; ++i) {
    const float mu1 = m1[i], mu2 = m2[i];
    const float mu1s = mu1 * mu1, mu2s = mu2 * mu2, mu12 = mu1 * mu2;
    const float v11 = s11[i] - mu1s;
    const float v22 = s22[i] - mu2s;
    const float v12 = s12[i] - mu12;
    const float num = (2.0f * mu12 + K1) * (2.0f * v12 + K2);
    const float den = (mu1s + mu2s + K1) * (v11 + v22 + K2);
    part += num / den;
  }

  // Wave32 butterfly reduction, lane 0 writes the per-tile partial.
#pragma unroll
  for (int off = 16; off > 0; off >>= 1)
    part += __shfl_xor(part, off, 32);
  if (lane == 0) partials[b] = part;
}

// Deterministic fixed-tree reduction of the 32768 tile partials.
__global__ __launch_bounds__(1024) void ssim_reduce_kernel(
    const float* __restrict__ partials, float* __restrict__ out, int n) {
  __shared__ float sm[1024];
  float s = 0.0f;
  for (int i = (int)threadIdx.x; i < n; i += 1024) s += partials[i];
  sm[threadIdx.x] = s;
  __syncthreads();
  for (int k = 512; k > 0; k >>= 1) {
    if ((int)threadIdx.x < k) sm[threadIdx.x] += sm[threadIdx.x + k];
    __syncthreads();
  }
  if (threadIdx.x == 0)
    out[0] = 1.0f - sm[0] / (float)(IMG_N * IMG_H * IMG_W);
}

extern "C" void kernel_launch(void* const* d_in, const int* in_sizes, int n_in,
                              void* d_out, int out_size, void* d_ws, size_t ws_size,
                              hipStream_t stream) {
  (void)in_sizes; (void)n_in; (void)out_size; (void)ws_size;
  const float* img1 = (const float*)d_in[0];
  const float* img2 = (const float*)d_in[1];
  float* ws  = (float*)d_ws;    // NTILES floats = 128 KB of scratch
  float* out = (float*)d_out;

  ssim_tile_kernel<<<NTILES, 32, 0, stream>>>(img1, img2, ws);
  ssim_reduce_kernel<<<1, 1024, 0, stream>>>(ws, out, NTILES);
}